// SE3Transformer_16698832847172
// MI455X (gfx1250) — compile-verified
//
#include <hip/hip_runtime.h>
#include <math.h>

typedef float v2f __attribute__((ext_vector_type(2)));
typedef float v8f __attribute__((ext_vector_type(8)));

#define CH 1024  // edge-chunk for radial MLP staging

// =====================================================================
// Device kernels
// =====================================================================

// Generic strided GEMM via fp32 WMMA (16x16x4).
//   C[n,m] (opt +=) = act( sum_k A[n*an + k*ak] * W[k*wk + m*wm] + bias[m] )
// One wave (32 threads) computes one 16x32 output tile (two 16x16 WMMA
// accumulators sharing the A fragment -> half the A traffic, 2 independent
// WMMA dependency chains per wave for latency hiding).
// Masked loads are branchless so EXEC is all-ones at every WMMA.
__global__ __launch_bounds__(32) void se3_gemm(
    const float* __restrict__ A, long an, long ak,
    const float* __restrict__ W, long wk, long wm,
    const float* __restrict__ bias,
    float* __restrict__ C, long cn, long cm,
    int N, int K, int M, int act, int acc)
{
  const int lane = threadIdx.x;
  const int half = lane >> 4;     // 0: K {0,1} | 1: K {2,3}  (A/B layout)
  const int l    = lane & 15;
  const int tn = blockIdx.x * 16;
  const int tm = blockIdx.y * 32;
  const int n  = tn + l;
  const int m0 = tm + l;
  const int m1 = tm + 16 + l;
  const long nc  = (n  < N) ? n  : (N - 1);
  const long mc0 = (m0 < M) ? m0 : (M - 1);
  const long mc1 = (m1 < M) ? m1 : (M - 1);
  const float nmask  = (n  < N) ? 1.f : 0.f;
  const float mmask0 = (m0 < M) ? 1.f : 0.f;
  const float mmask1 = (m1 < M) ? 1.f : 0.f;
  v8f c0 = {0.f,0.f,0.f,0.f,0.f,0.f,0.f,0.f};
  v8f c1 = {0.f,0.f,0.f,0.f,0.f,0.f,0.f,0.f};
#pragma unroll 2
  for (int k0 = 0; k0 < K; k0 += 4) {
    int ka = k0 + 2 * half;
    long ka0 = (ka < K) ? ka : (K - 1);
    long ka1 = (ka + 1 < K) ? (ka + 1) : (K - 1);
    float km0 = (ka     < K) ? 1.f : 0.f;
    float km1 = (ka + 1 < K) ? 1.f : 0.f;
    v2f a, b0, b1;
    a.x  = A[nc * an + ka0 * ak] * (nmask * km0);  // A 16x4: lanes0-15 K{0,1}, 16-31 K{2,3}
    a.y  = A[nc * an + ka1 * ak] * (nmask * km1);
    b0.x = W[ka0 * wk + mc0 * wm] * (mmask0 * km0);
    b0.y = W[ka1 * wk + mc0 * wm] * (mmask0 * km1);
    b1.x = W[ka0 * wk + mc1 * wm] * (mmask1 * km0);
    b1.y = W[ka1 * wk + mc1 * wm] * (mmask1 * km1);
    c0 = __builtin_amdgcn_wmma_f32_16x16x4_f32(false, a, false, b0, (short)0, c0,
                                               false, false);
    c1 = __builtin_amdgcn_wmma_f32_16x16x4_f32(false, a, false, b1, (short)0, c1,
                                               false, false);
  }
  // epilogue: D vgpr v holds row v (+8 for hi half), col = l
  if (m0 < M) {
    float bv = bias ? bias[m0] : 0.f;
#pragma unroll
    for (int v = 0; v < 8; ++v) {
      int nn = tn + v + 8 * half;
      if (nn < N) {
        long idx = (long)nn * cn + (long)m0 * cm;
        float val = c0[v];
        if (acc) val += C[idx];
        val += bv;
        if (act == 1) val = fmaxf(val, 0.f);
        else if (act == 2) val = (val > 0.f) ? val : (expf(val) - 1.f);
        C[idx] = val;
      }
    }
  }
  if (m1 < M) {
    float bv = bias ? bias[m1] : 0.f;
#pragma unroll
    for (int v = 0; v < 8; ++v) {
      int nn = tn + v + 8 * half;
      if (nn < N) {
        long idx = (long)nn * cn + (long)m1 * cm;
        float val = c1[v];
        if (acc) val += C[idx];
        val += bv;
        if (act == 1) val = fmaxf(val, 0.f);
        else if (act == 2) val = (val > 0.f) ? val : (expf(val) - 1.f);
        C[idx] = val;
      }
    }
  }
}

__global__ void se3_fill(float* p, float v, long n) {
  long i = (long)blockIdx.x * blockDim.x + threadIdx.x;
  if (i < n) p[i] = v;
}

// Per-edge basis: ef = [efeat, r], Y1, and b11[e][m][n][f] (K0,K1,K2 stacked)
__global__ void se3_basis(const float* __restrict__ xyz,
                          const int* __restrict__ src, const int* __restrict__ dst,
                          const float* __restrict__ efeat,
                          float* __restrict__ ef, float* __restrict__ y1,
                          float* __restrict__ b11, int E)
{
  int e = blockIdx.x * blockDim.x + threadIdx.x;
  if (e >= E) return;
  int s = src[e], d = dst[e];
  float dx = xyz[d*3+0]-xyz[s*3+0];
  float dy = xyz[d*3+1]-xyz[s*3+1];
  float dz = xyz[d*3+2]-xyz[s*3+2];
  float r = sqrtf(dx*dx + dy*dy + dz*dz + 1e-8f);
  float inv = 1.f / r;
  float x = dx*inv, y = dy*inv, z = dz*inv;
  const float C1 = 0.4886025119029199f;
  float Y10 = C1*y, Y11 = C1*z, Y12 = C1*x;          // stack([y,z,x])
  float Y2[5];
  Y2[0] = 1.0925484305920792f * x * y;
  Y2[1] = 1.0925484305920792f * y * z;
  Y2[2] = 0.31539156525252005f * (3.f*z*z - 1.f);
  Y2[3] = 1.0925484305920792f * x * z;
  Y2[4] = 0.5462742152960396f * (x*x - y*y);
  ef[e*3+0] = efeat[e*2+0];
  ef[e*3+1] = efeat[e*2+1];
  ef[e*3+2] = r;
  y1[e*3+0] = Y10; y1[e*3+1] = Y11; y1[e*3+2] = Y12;
  const float K0d = 0.28209479177387814f * 0.5773502691896258f;
  const float IS2 = 0.70710678118654752f, IS6 = 0.40824829046386302f;
  float K1[3][3] = {{0}}, K2[3][3] = {{0}};
  // K1[a][b] = (1/sqrt2) * eps[c,a,b] Y1[c]
  K1[1][2] =  Y10*IS2;  K1[2][1] = -Y10*IS2;
  K1[2][0] =  Y11*IS2;  K1[0][2] = -Y11*IS2;
  K1[0][1] =  Y12*IS2;  K1[1][0] = -Y12*IS2;
  // K2[a][b] = Q2[m,a,b] Y2[m]
  K2[2][0] += Y2[0]*IS2;  K2[0][2] += Y2[0]*IS2;
  K2[0][1] += Y2[1]*IS2;  K2[1][0] += Y2[1]*IS2;
  K2[1][1] += 2.f*Y2[2]*IS6; K2[0][0] -= Y2[2]*IS6; K2[2][2] -= Y2[2]*IS6;
  K2[2][1] += Y2[3]*IS2;  K2[1][2] += Y2[3]*IS2;
  K2[2][2] += Y2[4]*IS2;  K2[0][0] -= Y2[4]*IS2;
  float* B = b11 + (size_t)e * 27;
#pragma unroll
  for (int m = 0; m < 3; ++m)
#pragma unroll
    for (int n = 0; n < 3; ++n) {
      B[(m*3+n)*3+0] = (m == n) ? K0d : 0.f;
      B[(m*3+n)*3+1] = K1[m][n];
      B[(m*3+n)*3+2] = K2[m][n];
    }
}

// Apply one (di,do) pair kernel: out[e,o,(m)] += sum_{i,n,f} R[e,o,i,f]*basis*h[src]
__global__ void se3_pair_apply(const float* __restrict__ R,
                               const float* __restrict__ h,
                               const int* __restrict__ src,
                               const float* __restrict__ y1,
                               const float* __restrict__ b11,
                               float* __restrict__ out,
                               int e0, int cnt, int ci, int co, int di, int dd, int nf)
{
  int idx = blockIdx.x * blockDim.x + threadIdx.x;
  if (idx >= cnt * co) return;
  int el = idx / co, o = idx % co;
  int eg = e0 + el;
  int s = src[eg];
  const float* Re = R + ((size_t)el * co + o) * ci * nf;
  if (di == 0) {
    const float* hs = h + (size_t)s * ci;
    float t = 0.f;
    for (int i = 0; i < ci; ++i) t += Re[i] * hs[i];
    if (dd == 0) {
      out[(size_t)eg * co + o] += 0.28209479177387814f * t;   // b00 = Y0C
    } else {
      const float* Y = y1 + (size_t)eg * 3;
      float* op = out + ((size_t)eg * co + o) * 3;
#pragma unroll
      for (int m = 0; m < 3; ++m) op[m] += Y[m] * t;          // b01 = Y1[m]
    }
  } else {
    const float* hs = h + (size_t)s * ci * 3;
    if (dd == 0) {                                            // b10 = Y1[n]
      const float* Y = y1 + (size_t)eg * 3;
      float t = 0.f;
      for (int i = 0; i < ci; ++i) {
        float g = Y[0]*hs[i*3+0] + Y[1]*hs[i*3+1] + Y[2]*hs[i*3+2];
        t += Re[i] * g;
      }
      out[(size_t)eg * co + o] += t;
    } else {                                                  // nf==3, b11 full
      float T[3][3];
#pragma unroll
      for (int n = 0; n < 3; ++n)
#pragma unroll
        for (int f = 0; f < 3; ++f) T[n][f] = 0.f;
      for (int i = 0; i < ci; ++i) {
        float h0 = hs[i*3+0], h1 = hs[i*3+1], h2 = hs[i*3+2];
        const float* Ri = Re + i * 3;
#pragma unroll
        for (int f = 0; f < 3; ++f) {
          float r = Ri[f];
          T[0][f] += r * h0; T[1][f] += r * h1; T[2][f] += r * h2;
        }
      }
      const float* B = b11 + (size_t)eg * 27;
      float* op = out + ((size_t)eg * co + o) * 3;
#pragma unroll
      for (int m = 0; m < 3; ++m) {
        float a = 0.f;
#pragma unroll
        for (int n = 0; n < 3; ++n)
#pragma unroll
          for (int f = 0; f < 3; ++f) a += B[(m*3+n)*3+f] * T[n][f];
        op[m] += a;
      }
    }
  }
}

// Attention logits: logits[e,h] = scale * sum_{d,j,m} k_d[e,h*8+j,m] q_d[dst,h*8+j,m]
__global__ void se3_logits(const float* __restrict__ k0, const float* __restrict__ k1,
                           const float* __restrict__ q0, const float* __restrict__ q1,
                           const int* __restrict__ dst, float* __restrict__ logit,
                           int E, int has1, float scale)
{
  int e = blockIdx.x * blockDim.x + threadIdx.x;
  if (e >= E) return;
  int d = dst[e];
#pragma unroll
  for (int h = 0; h < 2; ++h) {
    float acc = 0.f;
    for (int j = 0; j < 8; ++j) {
      int c = h * 8 + j;
      acc += k0[(size_t)e*16 + c] * q0[(size_t)d*16 + c];
      if (has1)
#pragma unroll
        for (int m = 0; m < 3; ++m)
          acc += k1[((size_t)e*16 + c)*3 + m] * q1[((size_t)d*16 + c)*3 + m];
    }
    logit[(size_t)e*2 + h] = acc * scale;
  }
}

__device__ inline void atomicMaxF(float* addr, float v) {
  unsigned int* ua = (unsigned int*)addr;
  unsigned int old = *ua;
  while (true) {
    float f = __uint_as_float(old);
    if (f >= v) break;
    unsigned int assumed = old;
    old = atomicCAS(ua, assumed, __float_as_uint(v));
    if (old == assumed) break;
  }
}

__global__ void se3_segmax(const float* __restrict__ logit, const int* __restrict__ dst,
                           float* __restrict__ smax, int E)
{
  int i = blockIdx.x * blockDim.x + threadIdx.x;
  if (i >= E * 2) return;
  int e = i >> 1, h = i & 1;
  atomicMaxF(&smax[(size_t)dst[e]*2 + h], logit[i]);
}

__global__ void se3_expsum(const float* __restrict__ logit, const int* __restrict__ dst,
                           const float* __restrict__ smax, float* __restrict__ a,
                           float* __restrict__ ssum, int E)
{
  int i = blockIdx.x * blockDim.x + threadIdx.x;
  if (i >= E * 2) return;
  int e = i >> 1, h = i & 1;
  float v = expf(logit[i] - smax[(size_t)dst[e]*2 + h]);
  a[i] = v;
  atomicAdd(&ssum[(size_t)dst[e]*2 + h], v);
}

__global__ void se3_anorm(float* __restrict__ a, const int* __restrict__ dst,
                          const float* __restrict__ ssum, int E)
{
  int i = blockIdx.x * blockDim.x + threadIdx.x;
  if (i >= E * 2) return;
  int e = i >> 1, h = i & 1;
  a[i] /= (ssum[(size_t)dst[e]*2 + h] + 1e-9f);
}

// out[dst[e], c, m] += v[e,c,m] * a[e, head(c)]     (cv=16, heads=2)
__global__ void se3_wagg(const float* __restrict__ v, const float* __restrict__ a,
                         const int* __restrict__ dst, float* __restrict__ out,
                         int E, int md)
{
  long i = (long)blockIdx.x * blockDim.x + threadIdx.x;
  long tot = (long)E * 16 * md;
  if (i >= tot) return;
  int e = (int)(i / (16 * md));
  int rem = (int)(i % (16 * md));
  int c = rem / md, m = rem % md;
  float w = a[(size_t)e*2 + (c >> 3)];
  atomicAdd(&out[((size_t)dst[e]*16 + c)*md + m], v[i] * w);
}

// agg[dst[e], c] += msg[e, c]
__global__ void se3_scatter(const float* __restrict__ msg, const int* __restrict__ dst,
                            float* __restrict__ agg, int E, int co)
{
  long i = (long)blockIdx.x * blockDim.x + threadIdx.x;
  if (i >= (long)E * co) return;
  int e = (int)(i / co), c = (int)(i % co);
  atomicAdd(&agg[(size_t)dst[e]*co + c], msg[i]);
}

// Equivariant norm (in place): C=32 channels, md components per channel
__global__ void se3_gnorm(float* __restrict__ h, const float* __restrict__ g,
                          const float* __restrict__ b, int N, int md)
{
  int n = blockIdx.x * blockDim.x + threadIdx.x;
  if (n >= N) return;
  const int C = 32;
  float nrm[32];
  float* X = h + (size_t)n * C * md;
  float mu = 0.f;
  for (int c = 0; c < C; ++c) {
    float s = 0.f;
    for (int m = 0; m < md; ++m) { float v = X[c*md + m]; s += v*v; }
    nrm[c] = sqrtf(s + 1e-12f);
    mu += nrm[c];
  }
  mu /= C;
  float var = 0.f;
  for (int c = 0; c < C; ++c) { float d = nrm[c] - mu; var += d*d; }
  var /= C;
  float inv = 1.f / sqrtf(var + 1e-5f);
  for (int c = 0; c < C; ++c) {
    float ln = (nrm[c] - mu) * inv * g[c] + b[c];
    float sc = ((ln > 0.f) ? ln : 0.f) / nrm[c];
    for (int m = 0; m < md; ++m) X[c*md + m] *= sc;
  }
}

__global__ void se3_concat(const float* __restrict__ a, const float* __restrict__ b,
                           float* __restrict__ o, int N)
{
  int i = blockIdx.x * blockDim.x + threadIdx.x;
  if (i >= N * 64) return;
  int n = i >> 6, j = i & 63;
  o[i] = (j < 32) ? a[(size_t)n*32 + j] : b[(size_t)n*32 + (j - 32)];
}

__global__ void se3_final(const float* __restrict__ g, const float* __restrict__ w,
                          float* __restrict__ out, int K)
{
  if (blockIdx.x == 0 && threadIdx.x == 0) {
    float s = 0.f;
    for (int j = 0; j < K; ++j) { s += w[j] * g[j]; out[1 + j] = g[j]; }
    out[0] = s / (float)K;
  }
}

// =====================================================================
// Host-side parameter walking (jax pytree flatten: dicts sorted by key)
// =====================================================================

namespace {

struct PW { void* const* in; int i; const float* nx() { return (const float*)in[i++]; } };
struct Lin { const float* b; const float* w; };
static Lin getlin(PW& p) { Lin L; L.b = p.nx(); L.w = p.nx(); return L; }
struct Rad { Lin l1, l2, l3; };
static Rad getrad(PW& p) { Rad r; r.l1 = getlin(p); r.l2 = getlin(p); r.l3 = getlin(p); return r; }

struct ResL {
  Rad k00, k01, k10, k11, v00, v01, v10, v11;
  const float *q0, *q1, *proj0, *proj1;
  int ci0, ci1;    // ci1 < 0: no degree-1 input
};
// res_layer key order: 'k','proj','q','v'; pairs key order '0,0','0,1','1,0','1,1'
static ResL getres(PW& p, int ci0, int ci1) {
  ResL L; L.ci0 = ci0; L.ci1 = ci1;
  if (ci1 < 0) { L.k00 = getrad(p); }
  else { L.k00 = getrad(p); L.k01 = getrad(p); L.k10 = getrad(p); L.k11 = getrad(p); }
  L.proj0 = p.nx(); L.proj1 = p.nx();
  L.q0 = p.nx(); L.q1 = (ci1 >= 0) ? p.nx() : nullptr;
  if (ci1 < 0) { L.v00 = getrad(p); L.v01 = getrad(p); }
  else { L.v00 = getrad(p); L.v01 = getrad(p); L.v10 = getrad(p); L.v11 = getrad(p); }
  return L;
}

struct NormL { const float *b0, *g0, *b1, *g1; };
static NormL getnorm(PW& p) { NormL n; n.b0=p.nx(); n.g0=p.nx(); n.b1=p.nx(); n.g1=p.nx(); return n; }

struct Block {
  Rad c00, c10; const float* cself;
  NormL nrm[2];
  ResL L[2];
};
// block key order: 'conv'('pair':{'0,0','1,0'},'self':{'0'}), 'norm'[2], 'res'[2]
static Block getblock(PW& p, int ci0_l1, int ci1_l1) {
  Block B;
  B.c00 = getrad(p); B.c10 = getrad(p); B.cself = p.nx();
  B.nrm[0] = getnorm(p); B.nrm[1] = getnorm(p);
  B.L[0] = getres(p, ci0_l1, ci1_l1);
  B.L[1] = getres(p, 32, 32);
  return B;
}

struct WS {
  float *ef, *y1, *b11;
  float *hX0, *hX1, *hY0, *hY1;
  float *v0, *v1, *kk0, *kk1, *q0, *q1;
  float *logit, *attn, *smax, *ssum, *o0, *o1;
  float *t1, *t2, *Rb, *msg;
};

} // namespace

// =====================================================================
// kernel_launch
// =====================================================================

extern "C" void kernel_launch(void* const* d_in, const int* in_sizes, int n_in,
                              void* d_out, int out_size, void* d_ws, size_t ws_size,
                              hipStream_t stream)
{
  (void)n_in; (void)out_size; (void)ws_size;
  hipStream_t st = stream;

  // ---- unpack params (top-level params keys sorted) ----
  PW p{d_in, 0};
  Block Batm = getblock(p, 32, 1);     // 'atm' : fin=[(32,0),(1,1)]
  Block Bbnd = getblock(p, 32, -1);    // 'bnd' : fin=[(32,0)]
  Lin lin1_atm = getlin(p), lin1_bnd = getlin(p), lin1_res = getlin(p);
  Lin lin2_atm = getlin(p), lin2_bnd = getlin(p), lin2_res = getlin(p);
  Lin pa1 = getlin(p), pa2 = getlin(p);
  Block Bres = getblock(p, 32, -1);    // 'res'
  Lin w1 = getlin(p), w2 = getlin(p);

  // ---- remaining inputs in setup_inputs() order ----
  int bi = p.i;
  const float* feat_bnd  = (const float*)d_in[bi+0];
  const float* feat_res  = (const float*)d_in[bi+1];
  const float* l1_atm    = (const float*)d_in[bi+2];
  const float* xyz_atm   = (const float*)d_in[bi+3];
  const float* xyz_res   = (const float*)d_in[bi+4];
  const float* efeat_bnd = (const float*)d_in[bi+5];
  const float* efeat_res = (const float*)d_in[bi+6];
  const float* efeat_atm = (const float*)d_in[bi+7];
  const int*   eb        = (const int*)d_in[bi+8];
  const int*   er        = (const int*)d_in[bi+9];
  const int*   ea        = (const int*)d_in[bi+10];
  const float* r2a       = (const float*)d_in[bi+11];
  const float* lig       = (const float*)d_in[bi+12];

  const int Na = in_sizes[bi+3] / 3;
  const int Nr = in_sizes[bi+4] / 3;
  const int Eb = in_sizes[bi+5] / 2;
  const int Er = in_sizes[bi+6] / 2;
  const int Ea = in_sizes[bi+7] / 2;
  const int Klig = in_sizes[bi+12] / Na;

  const int EM = (Eb > Er ? (Eb > Ea ? Eb : Ea) : (Er > Ea ? Er : Ea));
  const int NM = (Na > Nr ? Na : Nr);

  // ---- workspace layout ----
  size_t off = 0;
  float* base = (float*)d_ws;
  auto alloc = [&](size_t n) { float* q = base + off; off += n; return q; };
  WS w;
  w.ef  = alloc((size_t)EM * 3);
  w.y1  = alloc((size_t)EM * 3);
  w.b11 = alloc((size_t)EM * 27);
  w.hX0 = alloc((size_t)NM * 32);  w.hX1 = alloc((size_t)NM * 96);
  w.hY0 = alloc((size_t)NM * 32);  w.hY1 = alloc((size_t)NM * 96);
  w.v0  = alloc((size_t)EM * 16);  w.v1  = alloc((size_t)EM * 48);
  w.kk0 = alloc((size_t)EM * 16);  w.kk1 = alloc((size_t)EM * 48);
  w.q0  = alloc((size_t)NM * 16);  w.q1  = alloc((size_t)NM * 48);
  w.logit = alloc((size_t)EM * 2); w.attn = alloc((size_t)EM * 2);
  w.smax  = alloc((size_t)NM * 2); w.ssum = alloc((size_t)NM * 2);
  w.o0  = alloc((size_t)NM * 16);  w.o1  = alloc((size_t)NM * 48);
  w.t1  = alloc((size_t)CH * 32);  w.t2  = alloc((size_t)CH * 32);
  w.Rb  = alloc((size_t)CH * 2048);
  w.msg = alloc((size_t)EM * 64);
  float* agg_bnd = alloc((size_t)Na * 32);
  float* agg_res = alloc((size_t)Nr * 32);
  float* agg_atm = alloc((size_t)Na * 64);
  float* bigA    = alloc((size_t)NM * 95);
  float* h0_b    = alloc((size_t)Na * 32);
  float* h0_r    = alloc((size_t)Nr * 32);
  float* hr2a    = alloc((size_t)Na * 32);
  float* catb    = alloc((size_t)Na * 64);
  float* c64     = alloc((size_t)Na * 64);
  float* h0_a    = alloc((size_t)Na * 32);
  float* tbuf    = alloc((size_t)Na * 64);
  float* gp      = alloc((size_t)Na);
  float* wp      = alloc((size_t)Na);
  float* gsel    = alloc(64);
  float* wsel    = alloc(64);

  // ---- host helpers ----
  auto fill = [&](float* ptr, float v, long n) {
    se3_fill<<<dim3((unsigned)((n + 255) / 256)), 256, 0, st>>>(ptr, v, n);
  };
  auto gemm = [&](const float* A, long an, long ak, const float* W, long wk, long wm,
                  const float* bias, float* C, long cn, long cm,
                  int N, int K, int M, int act, int acc) {
    dim3 g((N + 15) / 16, (M + 31) / 32);
    se3_gemm<<<g, 32, 0, st>>>(A, an, ak, W, wk, wm, bias, C, cn, cm, N, K, M, act, acc);
  };
  // out[n,o,m] (+)= sum_k A[n,k,m] * Wmat[o, koff+k]   (einsum 'oc,ncm->nom')
  auto mixm = [&](const float* A, int K, int md, const float* Wmat, int wld, int koff,
                  int M, float* out, int N, int acc) {
    for (int m = 0; m < md; ++m)
      gemm(A + m, (long)K * md, md, Wmat + koff, 1, wld, nullptr,
           out + m, (long)M * md, md, N, K, M, 0, acc);
  };
  // One (di,dd) pair-kernel contribution accumulated into `out`
  auto edge_pair = [&](const Rad& rp, const float* h, const int* src,
                       float* out, int E, int ci, int co, int di, int dd) {
    int nf = (di == 1 && dd == 1) ? 3 : 1;
    int M3 = co * ci * nf;
    for (int e0 = 0; e0 < E; e0 += CH) {
      int cnt = (E - e0 < CH) ? (E - e0) : CH;
      gemm(w.ef + (size_t)e0 * 3, 3, 1, rp.l1.w, 32, 1, rp.l1.b, w.t1, 32, 1, cnt, 3, 32, 1, 0);
      gemm(w.t1, 32, 1, rp.l2.w, 32, 1, rp.l2.b, w.t2, 32, 1, cnt, 32, 32, 1, 0);
      gemm(w.t2, 32, 1, rp.l3.w, M3, 1, rp.l3.b, w.Rb, M3, 1, cnt, 32, M3, 0, 0);
      int tot = cnt * co;
      se3_pair_apply<<<(tot + 127) / 128, 128, 0, st>>>(
          w.Rb, h, src, w.y1, w.b11, out, e0, cnt, ci, co, di, dd, nf);
    }
  };

  auto run_layer = [&](const ResL& L, const NormL& nm,
                       const float* h0, int c0, const float* h1, int c1,
                       const int* src, const int* dstp, int N, int E,
                       float* hn0, float* hn1) {
    // values
    fill(w.v0, 0.f, (long)E * 16);
    fill(w.v1, 0.f, (long)E * 48);
    edge_pair(L.v00, h0, src, w.v0, E, c0, 16, 0, 0);
    edge_pair(L.v01, h0, src, w.v1, E, c0, 16, 0, 1);
    if (c1 > 0) {
      edge_pair(L.v10, h1, src, w.v0, E, c1, 16, 1, 0);
      edge_pair(L.v11, h1, src, w.v1, E, c1, 16, 1, 1);
    }
    // keys
    fill(w.kk0, 0.f, (long)E * 16);
    edge_pair(L.k00, h0, src, w.kk0, E, c0, 16, 0, 0);
    if (c1 > 0) {
      edge_pair(L.k10, h1, src, w.kk0, E, c1, 16, 1, 0);
      fill(w.kk1, 0.f, (long)E * 48);
      edge_pair(L.k01, h0, src, w.kk1, E, c0, 16, 0, 1);
      edge_pair(L.k11, h1, src, w.kk1, E, c1, 16, 1, 1);
    }
    // queries
    mixm(h0, c0, 1, L.q0, c0, 0, 16, w.q0, N, 0);
    if (c1 > 0) mixm(h1, c1, 3, L.q1, c1, 0, 16, w.q1, N, 0);
    // logits + segment softmax over dst
    float dk = (c1 > 0) ? 32.f : 8.f;
    se3_logits<<<(E + 127) / 128, 128, 0, st>>>(w.kk0, w.kk1, w.q0, w.q1, dstp,
                                                w.logit, E, (c1 > 0) ? 1 : 0,
                                                1.f / sqrtf(dk));
    fill(w.smax, -1e30f, (long)N * 2);
    se3_segmax<<<(E * 2 + 255) / 256, 256, 0, st>>>(w.logit, dstp, w.smax, E);
    fill(w.ssum, 0.f, (long)N * 2);
    se3_expsum<<<(E * 2 + 255) / 256, 256, 0, st>>>(w.logit, dstp, w.smax, w.attn, w.ssum, E);
    se3_anorm<<<(E * 2 + 255) / 256, 256, 0, st>>>(w.attn, dstp, w.ssum, E);
    // weighted aggregation
    fill(w.o0, 0.f, (long)N * 16);
    fill(w.o1, 0.f, (long)N * 48);
    se3_wagg<<<(unsigned)(((long)E*16 + 255) / 256), 256, 0, st>>>(w.v0, w.attn, dstp, w.o0, E, 1);
    se3_wagg<<<(unsigned)(((long)E*48 + 255) / 256), 256, 0, st>>>(w.v1, w.attn, dstp, w.o1, E, 3);
    // projection (concat [agg, h])
    mixm(w.o0, 16, 1, L.proj0, 16 + c0, 0, 32, hn0, N, 0);
    mixm(h0, c0, 1, L.proj0, 16 + c0, 16, 32, hn0, N, 1);
    int wld1 = 16 + ((c1 > 0) ? c1 : 0);
    mixm(w.o1, 16, 3, L.proj1, wld1, 0, 32, hn1, N, 0);
    if (c1 > 0) mixm(h1, c1, 3, L.proj1, wld1, 16, 32, hn1, N, 1);
    // equivariant norm (in place)
    se3_gnorm<<<(N + 127) / 128, 128, 0, st>>>(hn0, nm.g0, nm.b0, N, 1);
    se3_gnorm<<<(N + 127) / 128, 128, 0, st>>>(hn1, nm.g1, nm.b1, N, 3);
  };

  auto run_block = [&](const Block& B, const float* h0in, int c0, const float* h1in, int c1,
                       const int* src, const int* dstp, int N, int E,
                       float* agg, int co) {
    run_layer(B.L[0], B.nrm[0], h0in, c0, h1in, c1, src, dstp, N, E, w.hX0, w.hX1);
    run_layer(B.L[1], B.nrm[1], w.hX0, 32, w.hX1, 32, src, dstp, N, E, w.hY0, w.hY1);
    // final graph conv -> degree-0, co channels
    fill(w.msg, 0.f, (long)E * co);
    edge_pair(B.c00, w.hY0, src, w.msg, E, 32, co, 0, 0);
    edge_pair(B.c10, w.hY1, src, w.msg, E, 32, co, 1, 0);
    fill(agg, 0.f, (long)N * co);
    se3_scatter<<<(unsigned)(((long)E*co + 255) / 256), 256, 0, st>>>(w.msg, dstp, agg, E, co);
    mixm(w.hY0, 32, 1, B.cself, 32, 0, co, agg, N, 1);
  };

  // ================= forward pass =================

  // ---- bnd block ----
  se3_basis<<<(Eb + 127) / 128, 128, 0, st>>>(xyz_atm, eb, eb + Eb, efeat_bnd,
                                              w.ef, w.y1, w.b11, Eb);
  gemm(feat_bnd, 95, 1, lin1_bnd.w, 95, 1, lin1_bnd.b, bigA, 95, 1, Na, 95, 95, 2, 0);
  gemm(bigA, 95, 1, lin2_bnd.w, 32, 1, lin2_bnd.b, h0_b, 32, 1, Na, 95, 32, 0, 0);
  run_block(Bbnd, h0_b, 32, nullptr, 0, eb, eb + Eb, Na, Eb, agg_bnd, 32);

  // ---- res block ----
  se3_basis<<<(Er + 127) / 128, 128, 0, st>>>(xyz_res, er, er + Er, efeat_res,
                                              w.ef, w.y1, w.b11, Er);
  gemm(feat_res, 29, 1, lin1_res.w, 29, 1, lin1_res.b, bigA, 29, 1, Nr, 29, 29, 2, 0);
  gemm(bigA, 29, 1, lin2_res.w, 32, 1, lin2_res.b, h0_r, 32, 1, Nr, 29, 32, 0, 0);
  run_block(Bres, h0_r, 32, nullptr, 0, er, er + Er, Nr, Er, agg_res, 32);

  // ---- atm block input: concat(h_b, r2a @ h_r) -> lins ----
  gemm(r2a, Nr, 1, agg_res, 32, 1, nullptr, hr2a, 32, 1, Na, Nr, 32, 0, 0);
  se3_concat<<<(Na * 64 + 255) / 256, 256, 0, st>>>(agg_bnd, hr2a, catb, Na);
  gemm(catb, 64, 1, lin1_atm.w, 64, 1, lin1_atm.b, c64, 64, 1, Na, 64, 64, 2, 0);
  gemm(c64, 64, 1, lin2_atm.w, 32, 1, lin2_atm.b, h0_a, 32, 1, Na, 64, 32, 0, 0);

  se3_basis<<<(Ea + 127) / 128, 128, 0, st>>>(xyz_atm, ea, ea + Ea, efeat_atm,
                                              w.ef, w.y1, w.b11, Ea);
  run_block(Batm, h0_a, 32, l1_atm, 1, ea, ea + Ea, Na, Ea, agg_atm, 64);

  // ---- heads: g = pa2(relu(pa1(g0))), w = w2(relu(w1(g0))) ----
  gemm(agg_atm, 64, 1, pa1.w, 64, 1, pa1.b, tbuf, 64, 1, Na, 64, 64, 1, 0);
  gemm(tbuf, 64, 1, pa2.w, 1, 1, pa2.b, gp, 1, 1, Na, 64, 1, 0, 0);
  gemm(agg_atm, 64, 1, w1.w, 64, 1, w1.b, tbuf, 64, 1, Na, 64, 64, 1, 0);
  gemm(tbuf, 64, 1, w2.w, 1, 1, w2.b, wp, 1, 1, Na, 64, 1, 0, 0);

  // ligand selection: sel[j] = sum_n lig[n,j] * p[n]   (lig^T @ p via strided GEMM)
  gemm(lig, 1, Klig, gp, 1, 0, nullptr, gsel, 1, 1, Klig, Na, 1, 0, 0);
  gemm(lig, 1, Klig, wp, 1, 0, nullptr, wsel, 1, 1, Klig, Na, 1, 0, 0);

  se3_final<<<1, 32, 0, st>>>(gsel, wsel, (float*)d_out, Klig);
}